// NewTable_40896678592650
// MI455X (gfx1250) — compile-verified
//
#include <hip/hip_runtime.h>
#include <hip/hip_fp16.h>

#define TAB_N 259
#define NUM_POINTS 32
#define LUT_ENTRIES 65536
#define LUT_BYTES (LUT_ENTRIES * 2)

// Bit-exact emulation of reference fp32_to_fp16_floor:
// truncate fp32 mantissa to 10 bits, then cast to fp16 (RTNE cast matches
// numpy astype for the subnormal / overflow corner cases too).
__device__ __forceinline__ __half fp16_floor(float v) {
  unsigned u = __float_as_uint(v) & 0xFFFFE000u;
  return __float2half(__uint_as_float(u));
}

// Full bit-exact scalar evaluation of reference() for one fp16 bit pattern.
__device__ __forceinline__ unsigned short eval_one(unsigned short xb,
                                                   const __half* __restrict__ cut16,
                                                   const float* __restrict__ tab,
                                                   const float* __restrict__ ms) {
  float xf = __half2float(__ushort_as_half(xb));
  if (xf != xf) xf = 0.0f;                       // NaN -> 0
  float c0  = __half2float(cut16[0]);
  float c10 = __half2float(cut16[10]);
  xf = fminf(fmaxf(xf, c0), c10);                // clip (values fp16-exact)

  // ci = clip(min(searchsorted_right(cp32, x), 10) - 1, 0) ==
  //      #{ j in 1..9 : cp[j] <= x }   (cp[0] <= x always after clamp)
  int ci = 0;
#pragma unroll
  for (int j = 1; j <= 9; ++j)
    ci += (__half2float(cut16[j]) <= xf) ? 1 : 0;

  float cutv   = __half2float(cut16[ci]);
  float scalev = ms[ci];

  __half t1 = fp16_floor(xf - cutv);                       // add16(x, -cut[ci])
  __half t2 = fp16_floor(__half2float(t1) * scalev);       // mul16(., scale[ci])
  float tempf = __half2float(t2);
  float fidx  = floorf(tempf);
  int   idx   = (int)fidx;
  __half hidx = __float2half((float)idx);                  // exact, idx in [0,32]
  __half dec  = fp16_floor(tempf - __half2float(hidx));    // add16(temp, -idx16)

  int indices = (ci == 0) ? idx : (1 + (ci - 1) * NUM_POINTS + idx);
  if (indices < 0) indices = 0;
  if (indices > TAB_N - 1) indices = TAB_N - 1;
  int ir = indices + 1;
  if (ir > TAB_N - 1) ir = TAB_N - 1;

  float left  = tab[indices];
  float right = tab[ir];

  __half interval = fp16_floor(right - left);                               // add16(right,-left)
  __half prod     = fp16_floor(__half2float(interval) * __half2float(dec)); // mul16
  __half interp   = fp16_floor(__half2float(prod) + left);                  // add16(., left)

  __half res = (__half2float(dec) == 0.0f) ? __float2half(left) : interp;
  return __half_as_ushort(res);
}

// Kernel 1: enumerate all 65536 fp16 bit patterns -> LUT in workspace.
__global__ void build_lut(unsigned short* __restrict__ lut,
                          const __half* __restrict__ cut16,
                          const float* __restrict__ tab,
                          const float* __restrict__ ms) {
  int i = blockIdx.x * blockDim.x + threadIdx.x;   // 0 .. 65535
  lut[i] = eval_one((unsigned short)i, cut16, tab, ms);
}

// CDNA5 async global->LDS copy, 16B per lane, tracked by ASYNCcnt.
// GVS addressing: mem = saddr(SGPR64) + voff(VGPR32); LDS dest = vdst VGPR.
__device__ __forceinline__ void async_copy_b128(unsigned lds_addr,
                                                unsigned voff,
                                                const void* saddr) {
  asm volatile("global_load_async_to_lds_b128 %0, %1, %2"
               :
               : "v"(lds_addr), "v"(voff), "s"(saddr)
               : "memory");
}

__device__ __forceinline__ void wait_asynccnt0() {
  asm volatile("s_wait_asynccnt 0x0" ::: "memory");
}

// Kernel 2: streaming gather through a 128KB LDS-resident LUT.
// uint4 = 8 fp16 elements per lane per iteration.
__global__ __launch_bounds__(512, 1)
void apply_lut(const uint4* __restrict__ in, uint4* __restrict__ out,
               const uint4* __restrict__ lut_g, int nvec) {
  extern __shared__ __align__(16) unsigned char smem[];
  unsigned short* slut = (unsigned short*)smem;

  // Stage 128KB LUT into LDS with async DMA copies (no VGPR round-trip).
  // Generic LDS pointer truncated to 32 bits == LDS byte offset (ISA 10.2).
  unsigned lds_base = (unsigned)(size_t)(void*)smem;
  for (int i = threadIdx.x; i < LUT_BYTES / 16; i += blockDim.x) {
    unsigned off = (unsigned)i * 16u;
    async_copy_b128(lds_base + off, off, (const void*)lut_g);
  }
  wait_asynccnt0();
  __syncthreads();

  int stride = gridDim.x * blockDim.x;
#pragma unroll 2
  for (int v = blockIdx.x * blockDim.x + threadIdx.x; v < nvec; v += stride) {
    __builtin_prefetch(&in[v + stride], 0, 0);   // global_prefetch_b8, next tile
    uint4 p = in[v];
    uint4 r;
    unsigned a;
    a = p.x; r.x = (unsigned)slut[a & 0xffffu] | ((unsigned)slut[a >> 16] << 16);
    a = p.y; r.y = (unsigned)slut[a & 0xffffu] | ((unsigned)slut[a >> 16] << 16);
    a = p.z; r.z = (unsigned)slut[a & 0xffffu] | ((unsigned)slut[a >> 16] << 16);
    a = p.w; r.w = (unsigned)slut[a & 0xffffu] | ((unsigned)slut[a >> 16] << 16);
    out[v] = r;
  }
}

// Tail elements (n not multiple of 8) — gathers LUT from global (tiny, hot in L2).
__global__ void apply_tail(const unsigned short* __restrict__ in,
                           unsigned short* __restrict__ out,
                           const unsigned short* __restrict__ lut,
                           int start, int n) {
  int i = start + blockIdx.x * blockDim.x + threadIdx.x;
  if (i < n) out[i] = lut[in[i]];
}

// Fallback if workspace is too small for the LUT: direct bit-exact evaluation.
__global__ void direct_kernel(const unsigned short* __restrict__ in,
                              unsigned short* __restrict__ out,
                              const __half* __restrict__ cut16,
                              const float* __restrict__ tab,
                              const float* __restrict__ ms, int n) {
  int i = blockIdx.x * blockDim.x + threadIdx.x;
  if (i < n) out[i] = eval_one(in[i], cut16, tab, ms);
}

extern "C" void kernel_launch(void* const* d_in, const int* in_sizes, int n_in,
                              void* d_out, int out_size, void* d_ws, size_t ws_size,
                              hipStream_t stream) {
  const __half* x      = (const __half*)d_in[0];   // fp16, 2048*32768
  const __half* cut16  = (const __half*)d_in[1];   // fp16, 11
  const float*  tab    = (const float*)d_in[2];    // fp32, 259
  const float*  ms     = (const float*)d_in[3];    // fp32, 10
  int n = in_sizes[0];
  unsigned short* outp = (unsigned short*)d_out;

  if (ws_size >= (size_t)LUT_BYTES) {
    unsigned short* lut = (unsigned short*)d_ws;
    build_lut<<<LUT_ENTRIES / 256, 256, 0, stream>>>(lut, cut16, tab, ms);

    int nvec = n >> 3;                 // uint4 = 8 fp16 elements
    if (nvec > 0) {
      int blocks = 1024;
      int needed = (nvec + 511) / 512;
      if (blocks > needed) blocks = needed;
      apply_lut<<<blocks, 512, LUT_BYTES, stream>>>(
          (const uint4*)x, (uint4*)d_out, (const uint4*)lut, nvec);
    }
    int rem = n & 7;
    if (rem) {
      apply_tail<<<1, 64, 0, stream>>>((const unsigned short*)x, outp, lut,
                                       n - rem, n);
    }
  } else {
    int blocks = (n + 255) / 256;
    direct_kernel<<<blocks, 256, 0, stream>>>((const unsigned short*)x, outp,
                                              cut16, tab, ms, n);
  }
}